// PostProcess_44255343018797
// MI455X (gfx1250) — compile-verified
//
#include <hip/hip_runtime.h>
#include <hip/hip_bf16.h>
#include <stdint.h>

// Problem sizes (fixed by setup_inputs)
#define B_  128
#define Q_  3000
#define C_  3
#define G_  32
#define NK  4096          // padded key count (pow2 for bitonic sort)
#define NT  1024          // threads per sorting block (32 wave32s)
#define PER (NK / NT)     // keys per thread = 4

// LDS layout (bytes):
//   [0,            96000)  sbbox : 3000 x 8 f32 staged by TDM (lives whole kernel)
//   [96000,       128768)  skey  : 4096 x u64   (phase 3+)
//   [128768,      140768)  sscore: 3000 x f32   (phase 3+)
//   [96000,       132000)  slog  : 9000 x f32   (phases 1-2 only, aliases skey/sscore)
#define LDS_BBOX_OFF  0
#define LDS_KEY_OFF   96000
#define LDS_SCORE_OFF (96000 + NK * 8)
#define LDS_LOG_OFF   96000
#define LDS_TOTAL     (96000 + NK * 8 + Q_ * 4)   // 140768 B (<320KB/WGP)

// Flat float offsets of the concatenated outputs, in reference return order
#define OFF_BOXES   ((size_t)0)
#define OFF_SCORES  (OFF_BOXES  + (size_t)B_ * 2 * Q_ * 4)   // 3,072,000
#define OFF_LABELS  (OFF_SCORES + (size_t)B_ * 2 * Q_)       // 3,840,000
#define OFF_VALID   (OFF_LABELS + (size_t)B_ * 2 * Q_)       // 4,608,000
#define OFF_GB      (OFF_VALID  + (size_t)B_ * 2 * Q_)       // 5,376,000
#define OFF_GLAB    (OFF_GB     + (size_t)B_ * 2 * G_ * 4)   // 5,408,768
#define OFF_GVALID  (OFF_GLAB   + (size_t)B_ * 2 * G_)       // 5,416,960

typedef unsigned int v4u __attribute__((ext_vector_type(4)));
typedef unsigned int v8u __attribute__((ext_vector_type(8)));

// Issue one TDM descriptor: contiguous 1D copy of `nelem` 4-byte elements
// (nelem <= 65535 so it fits tile_dim0) from global to LDS. Wave-level op;
// caller must gate to a single wave and later s_wait_tensorcnt + barrier.
__device__ __forceinline__ void tdm_load_1d_f32(unsigned lds_byte_off,
                                                const void* gptr,
                                                unsigned nelem) {
    unsigned long long ga = (unsigned long long)(uintptr_t)gptr;
    v4u g0;
    g0[0] = 1u;                                      // count=1 (valid), user mode
    g0[1] = lds_byte_off;                            // lds_addr [63:32]
    g0[2] = (unsigned)ga;                            // global_addr[31:0]
    g0[3] = (unsigned)((ga >> 32) & 0x01FFFFFFu)     // global_addr[56:32]
          | (2u << 30);                              // type=2 ("image")
    v8u g1;
    g1[0] = (2u << 16);                              // wg_mask=0, data_size=2 (4B)
    g1[1] = (nelem & 0xFFFFu) << 16;                 // tensor_dim0[15:0] @ bits 63:48
    g1[2] = ((nelem >> 16) & 0xFFFFu)                // tensor_dim0[31:16]
          | (1u << 16);                              // tensor_dim1 = 1
    g1[3] = (nelem & 0xFFFFu) << 16;                 // tile_dim0 @ bits 127:112
    g1[4] = 1u;                                      // tile_dim1=1, tile_dim2=0
    g1[5] = nelem;                                   // tensor_dim0_stride[31:0]
    g1[6] = 0u;                                      // stride hi / dim1_stride lo
    g1[7] = 0u;                                      // dim1_stride hi
    // Tensor <=2D: groups 2/3 NULL -> 2-operand form.
    asm volatile("tensor_load_to_lds %0, %1" :: "s"(g0), "s"(g1) : "memory");
}

__device__ __forceinline__ void softmax3(float l0, float l1, float l2,
                                         float& p0, float& p1, float& p2) {
    float m  = fmaxf(l0, fmaxf(l1, l2));
    float e0 = __expf(l0 - m);
    float e1 = __expf(l1 - m);
    float e2 = __expf(l2 - m);
    float inv = 1.0f / (e0 + e1 + e2);
    p0 = e0 * inv; p1 = e1 * inv; p2 = e2 * inv;
}

// Emit one pair of cxcywh->xyxy boxes scaled by (w,h,w,h) and masked.
__device__ __forceinline__ void emit_box_pair(float* dst, const float* src8,
                                              float w, float h, float v0, float v1) {
    float4 a = *(const float4*)(src8);
    float4 b = *(const float4*)(src8 + 4);
    float4 o0, o1;
    o0.x = (a.x - 0.5f * a.z) * w * v0;
    o0.y = (a.y - 0.5f * a.w) * h * v0;
    o0.z = (a.x + 0.5f * a.z) * w * v0;
    o0.w = (a.y + 0.5f * a.w) * h * v0;
    o1.x = (b.x - 0.5f * b.z) * w * v1;
    o1.y = (b.y - 0.5f * b.w) * h * v1;
    o1.z = (b.x + 0.5f * b.z) * w * v1;
    o1.w = (b.y + 0.5f * b.w) * h * v1;
    ((float4*)dst)[0] = o0;
    ((float4*)dst)[1] = o1;
}

// One block per image. Phases:
//  0) wave0 issues TDM load of the image's 96 KB bbox tile into LDS
//     (DMA overlaps phases 1-4; waited only before emit)
//  1) async-copy this image's 36 KB logits into LDS (ASYNCcnt path)
//  2) build 46-bit sort keys + scores in registers
//  3) repurpose logits LDS: keys[4096] + scores[3000]
//  4) bitonic sort of the keys (single sort == reference's double argsort)
//  5) wait TENSORcnt, then emit boxes/scores/labels/valid from LDS bbox
extern "C" __global__ void __launch_bounds__(NT)
postprocess_sort_emit(const float* __restrict__ logits,
                      const float* __restrict__ bbox,
                      const int* __restrict__ tsizes,
                      float* __restrict__ out) {
    extern __shared__ unsigned char smem[];
    float* sbbox  = (float*)(smem + LDS_BBOX_OFF);
    unsigned long long* skey = (unsigned long long*)(smem + LDS_KEY_OFF);
    float* sscore = (float*)(smem + LDS_SCORE_OFF);
    float* slog   = (float*)(smem + LDS_LOG_OFF);

    const int b   = blockIdx.x;
    const int tid = threadIdx.x;

    // ---- phase 0: TDM DMA of bbox tile (wave 0 only; EXEC uniform per wave) ----
    if (tid < 32) {
        tdm_load_1d_f32((unsigned)(uintptr_t)(void*)sbbox,
                        bbox + (size_t)b * Q_ * 8,
                        (unsigned)(Q_ * 8));          // 24000 x 4B = 96 KB
    }

    // ---- phase 1: async global -> LDS staging of logits (ASYNCcnt path) ----
    {
        const unsigned long long gbase =
            (unsigned long long)(uintptr_t)(logits + (size_t)b * (Q_ * C_));
        const unsigned ldsbase = (unsigned)(uintptr_t)(void*)slog;   // LDS byte offset
        const int ndw2 = (Q_ * C_) / 2;                              // 4500 x b64
        for (int i = tid; i < ndw2; i += NT) {
            unsigned laddr = ldsbase + (unsigned)i * 8u;
            unsigned long long ga = gbase + (unsigned long long)i * 8ull;
            asm volatile("global_load_async_to_lds_b64 %0, %1, off"
                         :: "v"(laddr), "v"(ga) : "memory");
        }
        asm volatile("s_wait_asynccnt 0" ::: "memory");
    }
    __syncthreads();

    // ---- phase 2: keys & scores into registers ----
    unsigned long long rkey[PER];
    float rsc[PER];
    #pragma unroll
    for (int k = 0; k < PER; ++k) {
        int q = tid + k * NT;                 // q in [0, 4096)
        unsigned long long key;
        float sc = 0.0f;
        if (q < Q_) {
            float l0 = slog[q * 3 + 0];
            float l1 = slog[q * 3 + 1];
            float l2 = slog[q * 3 + 2];
            float p0, p1, p2;
            softmax3(l0, l1, l2, p0, p1, p2);
            float pmax = fmaxf(p0, fmaxf(p1, p2));
            bool keep = pmax > 0.5f;
            int label = (p0 >= p1) ? 0 : 1;   // argmax over first two classes, ties->0
            unsigned cat = keep ? (unsigned)label : 2u;
            // p0 in (0,1): positive float bits are monotonic; ~bits => descending order
            unsigned inv = (cat < 2u) ? ~__float_as_uint(p0) : 0u;
            key = ((unsigned long long)cat << 44) |
                  ((unsigned long long)inv << 12) |
                  (unsigned long long)(unsigned)q;
            sc = fmaxf(p0, p1);               // scores = prob[:, :-1].max(-1)
        } else {
            key = (3ull << 44) | (unsigned long long)(unsigned)q;  // pad: sorts last
        }
        rkey[k] = key;
        rsc[k]  = sc;
    }
    __syncthreads();   // all logits reads complete before LDS is repurposed

    // ---- phase 3: stash keys + scores into LDS ----
    #pragma unroll
    for (int k = 0; k < PER; ++k) {
        int q = tid + k * NT;
        skey[q] = rkey[k];
        if (q < Q_) sscore[q] = rsc[k];
    }
    __syncthreads();

    // ---- phase 4: bitonic sort ascending (keys unique => stable-equivalent) ----
    for (unsigned kk = 2; kk <= NK; kk <<= 1) {
        for (unsigned j = kk >> 1; j > 0; j >>= 1) {
            #pragma unroll
            for (int t = 0; t < PER; ++t) {
                unsigned i = (unsigned)tid + (unsigned)t * NT;
                unsigned ixj = i ^ j;
                if (ixj > i) {
                    unsigned long long a = skey[i];
                    unsigned long long c = skey[ixj];
                    bool up = ((i & kk) == 0);
                    if (up ? (a > c) : (a < c)) {
                        skey[i]   = c;
                        skey[ixj] = a;
                    }
                }
            }
            __syncthreads();
        }
    }

    // ---- phase 5: ensure TDM bbox tile landed, then emit ----
    if (tid < 32) {
        __builtin_amdgcn_s_wait_tensorcnt(0);
    }
    __syncthreads();

    const float w = (float)tsizes[b * 2 + 1];
    const float h = (float)tsizes[b * 2 + 0];
    float* oboxes = out + OFF_BOXES  + (size_t)b * 2 * Q_ * 4;
    float* oscore = out + OFF_SCORES + (size_t)b * 2 * Q_;
    float* olab   = out + OFF_LABELS + (size_t)b * 2 * Q_;
    float* ovalid = out + OFF_VALID  + (size_t)b * 2 * Q_;

    for (int j = tid; j < Q_; j += NT) {     // sorted slots [Q_, NK) are pads
        unsigned long long key = skey[j];
        unsigned q   = (unsigned)(key & 0xFFFull);
        unsigned cat = (unsigned)(key >> 44);
        float v0 = (cat < 2u)  ? 1.0f : 0.0f;   // i0 | i1
        float v1 = (cat == 0u) ? 1.0f : 0.0f;   // i0
        emit_box_pair(oboxes + (size_t)(2 * j) * 4, sbbox + (size_t)q * 8, w, h, v0, v1);
        float sc = sscore[q];
        oscore[2 * j + 0] = sc * v0;
        oscore[2 * j + 1] = sc * v1;
        olab[2 * j + 0] = v1;                   // where(is0,1,0) * valid
        olab[2 * j + 1] = 0.0f;
        ovalid[2 * j + 0] = v0;
        ovalid[2 * j + 1] = v1;
    }
}

// Ground truth: tiny elementwise pass, B*G = 4096 entries.
extern "C" __global__ void postprocess_gt(const float* __restrict__ gt_boxes,
                                          const int* __restrict__ gt_labels,
                                          const int* __restrict__ tsizes,
                                          float* __restrict__ out) {
    int idx = blockIdx.x * blockDim.x + threadIdx.x;
    if (idx >= B_ * G_) return;
    int b = idx / G_;
    int g = idx % G_;
    const float w = (float)tsizes[b * 2 + 1];
    const float h = (float)tsizes[b * 2 + 0];
    int l = gt_labels[idx];
    float v0 = (l == 0 || l == 1) ? 1.0f : 0.0f;
    float v1 = (l == 0) ? 1.0f : 0.0f;

    float* ogb = out + OFF_GB + (size_t)b * 2 * G_ * 4 + (size_t)(2 * g) * 4;
    emit_box_pair(ogb, gt_boxes + (size_t)idx * 8, w, h, v0, v1);

    size_t lo = (size_t)b * 2 * G_ + 2 * g;
    out[OFF_GLAB + lo + 0]   = v1;    // where(g0,1,0) * gvalid
    out[OFF_GLAB + lo + 1]   = 0.0f;
    out[OFF_GVALID + lo + 0] = v0;
    out[OFF_GVALID + lo + 1] = v1;
}

extern "C" void kernel_launch(void* const* d_in, const int* in_sizes, int n_in,
                              void* d_out, int out_size, void* d_ws, size_t ws_size,
                              hipStream_t stream) {
    const float* logits = (const float*)d_in[0];   // [B,Q,3]  f32
    const float* bbox   = (const float*)d_in[1];   // [B,Q,8]  f32
    const float* gtb    = (const float*)d_in[2];   // [B,G,8]  f32
    const int*   gtl    = (const int*)d_in[3];     // [B,G]    i32
    const int*   tsz    = (const int*)d_in[4];     // [B,2]    i32
    float* out = (float*)d_out;

    hipLaunchKernelGGL(postprocess_sort_emit, dim3(B_), dim3(NT), LDS_TOTAL, stream,
                       logits, bbox, tsz, out);
    hipLaunchKernelGGL(postprocess_gt, dim3((B_ * G_ + 255) / 256), dim3(256), 0, stream,
                       gtb, gtl, tsz, out);
}